// SurfaceLoss_82841329205896
// MI455X (gfx1250) — compile-verified
//
#include <hip/hip_runtime.h>
#include <float.h>

typedef __attribute__((ext_vector_type(2))) float v2f;
typedef __attribute__((ext_vector_type(8))) float v8f;

#define NPTS 16384
#define BK   1024                     // columns of B staged in LDS per chunk
#define INV_ND (1.0f / (16384.0f * 3.0f))

// ---------------------------------------------------------------------------
// matched = mean((x - y)^2): grid-stride reduction, atomicAdd into out[0]
// ---------------------------------------------------------------------------
__global__ __launch_bounds__(256)
void matched_loss_kernel(const float* __restrict__ x,
                         const float* __restrict__ y,
                         float* __restrict__ out, int n) {
    float acc = 0.0f;
    for (int i = blockIdx.x * blockDim.x + threadIdx.x; i < n;
         i += gridDim.x * blockDim.x) {
        float d = x[i] - y[i];
        acc = fmaf(d, d, acc);
    }
    // wave32 reduction
    #pragma unroll
    for (int off = 16; off > 0; off >>= 1)
        acc += __shfl_xor(acc, off, 32);

    __shared__ float smem[8];
    const int lane = threadIdx.x & 31;
    const int wv   = threadIdx.x >> 5;
    if (lane == 0) smem[wv] = acc;
    __syncthreads();
    if (threadIdx.x < 8) {
        float v = smem[threadIdx.x];
        #pragma unroll
        for (int off = 4; off > 0; off >>= 1)
            v += __shfl_xor(v, off, 8);
        if (threadIdx.x == 0) atomicAdd(out, v * INV_ND);
    }
}

// ---------------------------------------------------------------------------
// One chamfer direction: for each row a of A, min over all b in B of
// ||a-b||^2, summed over rows, scaled by 1/(N*D), atomicAdd into out[1].
//
// Homogeneous-coordinate WMMA (V_WMMA_F32_16X16X4_F32, K=4):
//   A row = (-2ax, -2ay, -2az, 1)
//   B col = ( bx,   by,   bz,  |b|^2)
//   =>  A x B (C = 0)  =  |b|^2 - 2 a.b   (= d2 minus the row-constant |a|^2)
//
// |a|^2 is loop-invariant under min_j, so it is added once per row in the
// epilogue:  min_j d2 = a2 + min_j (b2 - 2 a.b).  The inner loop is just:
//   ds_load_b64 (B operand halves straight from the float4 LDS tile)
//   v_wmma_f32_16x16x4_f32 with C = 0
//   8x fmin (fuses across unrolled tiles into v_min3_num_f32)
//
// LDS: column c stored as float4(bx,by,bz,b2); lanes 0-15 read dwords
// {4c,4c+1} = (bx,by), lanes 16-31 read {4c+2,4c+3} = (bz,b2): one
// branchless conflict-free ds_load_b64 per lane.
// ---------------------------------------------------------------------------
__global__ __launch_bounds__(256)
void chamfer_dir_kernel(const float* __restrict__ A,
                        const float* __restrict__ B,
                        float* __restrict__ out, int nB) {
    __shared__ float4 btile[BK];      // (bx, by, bz, |b|^2) per column : 16 KB

    const int tid  = threadIdx.x;
    const int lane = tid & 31;
    const int wv   = tid >> 5;
    const int n16  = lane & 15;       // column-within-tile / row-within-tile
    const int h    = lane >> 4;       // half-wave selector

    // ---- load this wave's 16 rows of A (fixed for whole kernel) ----
    const int row = blockIdx.x * 128 + wv * 16 + n16;
    const float ax = A[row * 3 + 0];
    const float ay = A[row * 3 + 1];
    const float az = A[row * 3 + 2];
    const float a2 = ax * ax + ay * ay + az * az;

    // A operand: lanes 0-15 hold (k0,k1)=(-2ax,-2ay), lanes 16-31 hold
    // (k2,k3)=(-2az, 1). Branchless value select, hoisted out of all loops.
    v2f aop;
    aop.x = h ? (-2.0f * az) : (-2.0f * ax);
    aop.y = h ? 1.0f         : (-2.0f * ay);

    float rmin[8];
    #pragma unroll
    for (int v = 0; v < 8; ++v) rmin[v] = FLT_MAX;

    const v2f* __restrict__ bvec = (const v2f*)btile;

    for (int chunk = 0; chunk < nB; chunk += BK) {
        __syncthreads();
        // ---- cooperative stage of BK columns of B into LDS ----
        for (int c = tid; c < BK; c += 256) {
            const int col = chunk + c;
            const float bx = B[col * 3 + 0];
            const float by = B[col * 3 + 1];
            const float bz = B[col * 3 + 2];
            btile[c] = make_float4(bx, by, bz, bx * bx + by * by + bz * bz);
        }
        __syncthreads();

        #pragma unroll 4
        for (int t = 0; t < BK / 16; ++t) {
            // one branchless ds_load_b64: this lane's half of the B operand
            const v2f bop = bvec[(t * 16 + n16) * 2 + h];

            v8f c = {};   // C = 0: WMMA output is (b2 - 2 a.b) for the tile
            c = __builtin_amdgcn_wmma_f32_16x16x4_f32(
                    false, aop, false, bop, (short)0, c, false, false);

            #pragma unroll
            for (int v = 0; v < 8; ++v) rmin[v] = fminf(rmin[v], c[v]);
        }
    }

    // ---- reduce: min across the 16 lanes sharing each row ----
    #pragma unroll
    for (int v = 0; v < 8; ++v) {
        float m = rmin[v];
        m = fminf(m, __shfl_xor(m, 1, 16));
        m = fminf(m, __shfl_xor(m, 2, 16));
        m = fminf(m, __shfl_xor(m, 4, 16));
        m = fminf(m, __shfl_xor(m, 8, 16));
        rmin[v] = m;
    }
    // sum of row-mins: half 0 holds rows 0-7, half 1 holds rows 8-15
    float s = 0.0f;
    #pragma unroll
    for (int v = 0; v < 8; ++v) s += rmin[v];

    // deferred |a|^2 term: sum a2 over the wave's 16 rows (lanes 0-15 hold
    // one row each; lanes 16-31 are duplicates within their width-16 group)
    float sa = a2;
    sa += __shfl_xor(sa, 1, 16);
    sa += __shfl_xor(sa, 2, 16);
    sa += __shfl_xor(sa, 4, 16);
    sa += __shfl_xor(sa, 8, 16);

    const float s0 = __shfl(s, 0, 32);    // rows 0-7  min-sum
    const float s1 = __shfl(s, 16, 32);   // rows 8-15 min-sum
    const float sA = __shfl(sa, 0, 32);   // sum of a2 over all 16 rows
    if (lane == 0) atomicAdd(out + 1, (s0 + s1 + sA) * INV_ND);
}

// ---------------------------------------------------------------------------
extern "C" void kernel_launch(void* const* d_in, const int* in_sizes, int n_in,
                              void* d_out, int out_size, void* d_ws, size_t ws_size,
                              hipStream_t stream) {
    const float* x = (const float*)d_in[0];   // [16384, 3]
    const float* y = (const float*)d_in[1];   // [16384, 3]
    float* out = (float*)d_out;               // [2]: matched, chamfer

    hipMemsetAsync(out, 0, 2 * sizeof(float), stream);

    matched_loss_kernel<<<48, 256, 0, stream>>>(x, y, out, NPTS * 3);

    // chamfer = (1/(N*D)) * [ sum_i min_j d2(x_i,y_j) + sum_j min_i d2(y_j,x_i) ]
    chamfer_dir_kernel<<<NPTS / 128, 256, 0, stream>>>(x, y, out, NPTS);
    chamfer_dir_kernel<<<NPTS / 128, 256, 0, stream>>>(y, x, out, NPTS);
}